// Conv2d_61615600828583
// MI455X (gfx1250) — compile-verified
//
#include <hip/hip_runtime.h>

// LUT-conv refactored as: out_t = a + b*x0 + c*x1 + d*x0*x1  (per-table WHT of LUT)
// Dense (a,b,c) part  -> im2col GEMM K=145 via V_WMMA_F32_16X16X4_F32 (exact fp32)
// Sparse (d*x0*x1)    -> VALU gathers from LDS-staged patch
// Both WMMA operands (patch A, weights B) are fed from LDS.

typedef __attribute__((ext_vector_type(2))) float v2f;
typedef __attribute__((ext_vector_type(8))) float v8f;

#define B_    8
#define IC_   16
#define H_    32
#define W_    32
#define OC_   32
#define TABLES_ 4608
#define RPOC_ 144      // tables per output channel
#define KDIM_ 145      // 144 patch slots + 1 bias
#define KPAD_ 148      // K rounded up to multiple of 4 for WMMA
#define PSTR_ 165      // LDS patch row stride (odd -> conflict-free A reads)
#define WSTR_ 40       // LDS W1 row stride (+80w per 2 rows = +16 banks -> conflict-free B reads)

// ---------------------------------------------------------------------------
// Build dense linear weight matrix W1[KPAD_][OC_] deterministically:
// W1[j][oc] = sum over this oc's tables of b (if j0==j) + c (if j1==j);
// W1[144][oc] = sum of a (bias row).  Rows 145..147 stay zero (K padding).
// ---------------------------------------------------------------------------
__global__ void build_w1_kernel(const float* __restrict__ w_luts,
                                const int* __restrict__ mc,
                                const int* __restrict__ mkh,
                                const int* __restrict__ mkw,
                                float* __restrict__ W1) {
    int tid = blockIdx.x * blockDim.x + threadIdx.x;
    if (tid >= KPAD_ * OC_) return;
    int oc = tid & (OC_ - 1);
    int j  = tid >> 5;
    float sum = 0.0f;
    if (j <= 144) {
        for (int r = 0; r < RPOC_; ++r) {
            int t = oc * RPOC_ + r;
            float w0 = w_luts[4 * t + 0];
            float w1 = w_luts[4 * t + 1];
            float w2 = w_luts[4 * t + 2];
            float w3 = w_luts[4 * t + 3];
            if (j == 144) {
                sum += 0.25f * (w0 + w1 + w2 + w3);          // a (bias)
            } else {
                int m0 = 2 * t, m1 = 2 * t + 1;
                int j0 = mc[m0] * 9 + mkh[m0] * 3 + mkw[m0];
                int j1 = mc[m1] * 9 + mkh[m1] * 3 + mkw[m1];
                if (j0 == j) sum += 0.25f * (-w0 + w1 - w2 + w3);  // b
                if (j1 == j) sum += 0.25f * (-w0 - w1 + w2 + w3);  // c
            }
        }
    }
    W1[j * OC_ + oc] = sum;
}

// ---------------------------------------------------------------------------
// Per-table bilinear coefficient d and packed patch indices (j0 | j1<<8)
// ---------------------------------------------------------------------------
__global__ void build_quad_kernel(const float* __restrict__ w_luts,
                                  const int* __restrict__ mc,
                                  const int* __restrict__ mkh,
                                  const int* __restrict__ mkw,
                                  float* __restrict__ qd,
                                  unsigned* __restrict__ qidx) {
    int t = blockIdx.x * blockDim.x + threadIdx.x;
    if (t >= TABLES_) return;
    float w0 = w_luts[4 * t + 0];
    float w1 = w_luts[4 * t + 1];
    float w2 = w_luts[4 * t + 2];
    float w3 = w_luts[4 * t + 3];
    int m0 = 2 * t, m1 = 2 * t + 1;
    int j0 = mc[m0] * 9 + mkh[m0] * 3 + mkw[m0];
    int j1 = mc[m1] * 9 + mkh[m1] * 3 + mkw[m1];
    qd[t]   = 0.25f * (w0 - w1 - w2 + w3);                  // d
    qidx[t] = (unsigned)j0 | ((unsigned)j1 << 8);
}

// ---------------------------------------------------------------------------
// Main fused kernel. One workgroup per output row (b, oh): 128 threads = 4
// wave32s; each wave owns one 16-position x 16-oc tile.
//   1) stage im2col patch p[32 pos][KPAD_] and W1 into LDS
//   2) dense part: 37x V_WMMA_F32_16X16X4_F32 over K=148, A & B from LDS
//   3) sparse part: 144 gathered d*x0*x1 FMAs per output on VALU
// ---------------------------------------------------------------------------
__global__ __launch_bounds__(128) void lutconv_main_kernel(
    const float* __restrict__ input,
    const float* __restrict__ W1,
    const float* __restrict__ qd,
    const unsigned* __restrict__ qidx,
    float* __restrict__ out) {

    __shared__ float p[32 * PSTR_];        // 21120 B
    __shared__ float w[KPAD_ * WSTR_];     // 23680 B

    const int row = blockIdx.x;            // 0..255
    const int b   = row >> 5;
    const int oh  = row & 31;
    const int tid = threadIdx.x;

    // ---- stage patch: 32 positions x 148 K-slots (uniform 37 trips/thread)
    for (int s = tid; s < 32 * KPAD_; s += 128) {
        int pos = s / KPAD_;               // ow
        int j   = s - pos * KPAD_;
        float v = 0.0f;
        if (j < 144) {
            int c   = j / 9;
            int rem = j - 9 * c;
            int kh  = rem / 3;
            int kw  = rem - 3 * kh;
            int ih  = oh + kh - 1;
            int iw  = pos + kw - 1;
            if ((unsigned)ih < (unsigned)H_ && (unsigned)iw < (unsigned)W_)
                v = input[((b * IC_ + c) * H_ + ih) * W_ + iw];
        } else if (j == 144) {
            v = 1.0f;                      // bias feature
        }
        p[pos * PSTR_ + j] = v;
    }
    // ---- stage W1 (row-major 32 -> LDS stride-40 rows), 37 trips/thread
    for (int s = tid; s < KPAD_ * OC_; s += 128) {
        int j  = s >> 5;
        int oc = s & (OC_ - 1);
        w[j * WSTR_ + oc] = W1[s];
    }
    __syncthreads();

    const int wave = tid >> 5;
    const int lane = tid & 31;
    const int half = lane >> 4;            // 0: K even pair, 1: K odd pair (A/B layout)
    const int idx  = lane & 15;            // M index for A, N index for B
    const int P0   = (wave >> 1) * 16;     // position-tile base (0 or 16)
    const int ocb  = (wave & 1) * 16;      // oc-tile base       (0 or 16)
    const int oc   = ocb + idx;

    v8f acc = {0.f, 0.f, 0.f, 0.f, 0.f, 0.f, 0.f, 0.f};

    // ---- dense linear + bias via WMMA f32 16x16x4 (K = 148 -> 37 steps)
    // A layout: lane(M=idx) holds K = k + 2*half + {0,1} in its 2 VGPRs
    // B layout: lane(N=idx) holds K = k + 2*half + {0,1} rows of W1
    const float* prow = &p[(P0 + idx) * PSTR_];
    const float* wcol = &w[2 * half * WSTR_ + oc];
    for (int k = 0; k < KPAD_; k += 4) {
        v2f a, bm;
        a.x  = prow[k + 2 * half];
        a.y  = prow[k + 2 * half + 1];
        bm.x = wcol[k * WSTR_];
        bm.y = wcol[k * WSTR_ + WSTR_];
        acc  = __builtin_amdgcn_wmma_f32_16x16x4_f32(
                   false, a, false, bm, (short)0, acc, false, false);
    }

    // ---- sparse bilinear part: acc[i] holds (M = i + 8*half, N = idx)
    const float*    qdp = &qd[oc * RPOC_];
    const unsigned* qip = &qidx[oc * RPOC_];
    const float* pbase  = &p[(P0 + 8 * half) * PSTR_];
    for (int r = 0; r < RPOC_; ++r) {
        float    d  = qdp[r];
        unsigned ij = qip[r];
        int j0 = (int)(ij & 255u);
        int j1 = (int)((ij >> 8) & 255u);
#pragma unroll
        for (int i = 0; i < 8; ++i) {
            float x0 = pbase[i * PSTR_ + j0];
            float x1 = pbase[i * PSTR_ + j1];
            acc[i] += d * x0 * x1;
        }
    }

    // ---- store: out[b][oc][oh][ow], ow = P0 + 8*half + i (8 consecutive)
    float* o = &out[(((b * OC_ + oc) * H_ + oh) * W_) + P0 + 8 * half];
#pragma unroll
    for (int i = 0; i < 8; ++i) o[i] = acc[i];
}

// ---------------------------------------------------------------------------
extern "C" void kernel_launch(void* const* d_in, const int* in_sizes, int n_in,
                              void* d_out, int out_size, void* d_ws, size_t ws_size,
                              hipStream_t stream) {
    const float* input  = (const float*)d_in[0];
    const float* w_luts = (const float*)d_in[1];
    const int*   mc     = (const int*)d_in[2];
    const int*   mkh    = (const int*)d_in[3];
    const int*   mkw    = (const int*)d_in[4];
    float* out = (float*)d_out;

    char* ws = (char*)d_ws;
    float*    W1 = (float*)ws;                              // 148*32 floats = 18944 B
    float*    qd = (float*)(ws + 18944);                    // 4608 floats   = 18432 B
    unsigned* qi = (unsigned*)(ws + 18944 + 18432);         // 4608 u32      = 18432 B
    // total workspace: 55808 B

    build_w1_kernel<<<(KPAD_ * OC_ + 127) / 128, 128, 0, stream>>>(w_luts, mc, mkh, mkw, W1);
    build_quad_kernel<<<(TABLES_ + 127) / 128, 128, 0, stream>>>(w_luts, mc, mkh, mkw, qd, qi);
    lutconv_main_kernel<<<B_ * H_, 128, 0, stream>>>(input, W1, qd, qi, out);
}